// GCN_16853451669954
// MI455X (gfx1250) — compile-verified
//
#include <hip/hip_runtime.h>

typedef __attribute__((ext_vector_type(2))) float v2f;
typedef __attribute__((ext_vector_type(8))) float v8f;

#define EPS 1e-5f

// ---------------------------------------------------------------- utilities
__global__ void k_fill_f32(float* p, float v, int n) {
  int i = blockIdx.x * blockDim.x + threadIdx.x;
  if (i < n) p[i] = v;
}

__global__ void k_fill_u32(unsigned* p, unsigned v, int n) {
  int i = blockIdx.x * blockDim.x + threadIdx.x;
  if (i < n) p[i] = v;
}

// ------------------------------------------------- degree / inv-sqrt-degree
__global__ void k_deg(const int* __restrict__ dst, float* __restrict__ deg, int E) {
  int i = blockIdx.x * blockDim.x + threadIdx.x;
  if (i < E) atomicAdd(&deg[dst[i]], 1.0f);
}

__global__ void k_rsqrt(const float* __restrict__ deg, float* __restrict__ dis, int n) {
  int i = blockIdx.x * blockDim.x + threadIdx.x;
  if (i < n) dis[i] = rsqrtf(deg[i]);
}

// ------------------------------------------------------------ WMMA f32 GEMM
// D[M,64] = A[M,K] * W[K,64].  M % 16 == 0, K % 4 == 0.  One wave per 16-row
// tile; four 16x16 accumulators along N; K stepped by 4 via
// V_WMMA_F32_16X16X4_F32 (full-precision f32 matrix pipe on CDNA5).
//
// Fragment layouts (ISA 7.12.2, 32-bit operands, wave32):
//   A 16x4 : lanes 0-15 -> M=lane; VGPR0 = K{0|2}, VGPR1 = K{1|3} by lane-half
//   B 4x16 : assumed symmetric: lanes -> N; VGPR0 = K{0|2}, VGPR1 = K{1|3}
//   C/D    : VGPR r: lanes 0-15 -> M=r, N=lane; lanes 16-31 -> M=8+r
__global__ void k_gemm_n64(const float* __restrict__ A,
                           const float* __restrict__ W,
                           float* __restrict__ D, int K) {
  const int m0   = blockIdx.x * 16;
  const int lane = threadIdx.x;           // 0..31
  const int half = lane >> 4;             // 0 | 1
  const int l    = lane & 15;

  v8f acc[4] = {};
  const float* arow = A + (size_t)(m0 + l) * K;

  for (int k0 = 0; k0 < K; k0 += 4) {
    v2f a;
    a.x = arow[k0 + 2 * half + 0];
    a.y = arow[k0 + 2 * half + 1];
#pragma unroll
    for (int t = 0; t < 4; ++t) {
      const int n0 = t * 16;
      v2f b;
      b.x = W[(k0 + 2 * half + 0) * 64 + n0 + l];
      b.y = W[(k0 + 2 * half + 1) * 64 + n0 + l];
      acc[t] = __builtin_amdgcn_wmma_f32_16x16x4_f32(
          /*neg_a=*/false, a, /*neg_b=*/false, b,
          /*c_mod=*/(short)0, acc[t], /*reuse_a=*/false, /*reuse_b=*/false);
    }
  }

#pragma unroll
  for (int t = 0; t < 4; ++t) {
    const int n0 = t * 16;
#pragma unroll
    for (int r = 0; r < 8; ++r)
      D[(size_t)(m0 + r + 8 * half) * 64 + n0 + l] = acc[t][r];
  }
}

// ------------------------------------- self-loop init:  agg = h*dis^2 + bias
__global__ void k_selfloop(const float* __restrict__ h,
                           const float* __restrict__ dis,
                           const float* __restrict__ bias,
                           float* __restrict__ agg, int n64) {
  int i = blockIdx.x * blockDim.x + threadIdx.x;
  if (i >= n64) return;
  int row = i >> 6, c = i & 63;
  float d = dis[row];
  agg[i] = h[i] * d * d + bias[c];
}

// -------------------- edge scatter:  agg[dst] += h[src] * dis[src]*dis[dst]
// One thread per (edge, 4-feature chunk); 16 consecutive threads cover one
// edge's 64-float row -> coalesced 256B gather, L2-resident f32 atomics.
__global__ void k_edge(const int* __restrict__ src, const int* __restrict__ dst,
                       const float* __restrict__ dis,
                       const float* __restrict__ h,
                       float* __restrict__ agg, int E) {
  int id = blockIdx.x * blockDim.x + threadIdx.x;
  int e = id >> 4;
  if (e >= E) return;
  int c0 = (id & 15) << 2;
  int s = src[e], d = dst[e];
  float nrm = dis[s] * dis[d];
  const float4 hv = *(const float4*)(h + (size_t)s * 64 + c0);
  float* ap = agg + (size_t)d * 64 + c0;
  atomicAdd(ap + 0, hv.x * nrm);
  atomicAdd(ap + 1, hv.y * nrm);
  atomicAdd(ap + 2, hv.z * nrm);
  atomicAdd(ap + 3, hv.w * nrm);
}

// -------------------------------------------- BatchNorm (64 features) stats
__global__ void k_bnstats(const float* __restrict__ x, int n,
                          float* __restrict__ sums /*[128]*/) {
  __shared__ float ssum[256], ssq[256];
  int t = threadIdx.x;
  int c = t & 63, rsub = t >> 6;
  float s = 0.f, q = 0.f;
  for (int r = blockIdx.x * 4 + rsub; r < n; r += gridDim.x * 4) {
    float v = x[(size_t)r * 64 + c];
    s += v;
    q += v * v;
  }
  ssum[t] = s; ssq[t] = q;
  __syncthreads();
  if (rsub == 0) {
    s = ssum[c] + ssum[64 + c] + ssum[128 + c] + ssum[192 + c];
    q = ssq[c] + ssq[64 + c] + ssq[128 + c] + ssq[192 + c];
    atomicAdd(&sums[c], s);
    atomicAdd(&sums[64 + c], q);
  }
}

// fold BN into per-feature scale/shift:  y = x*ksc[c] + ksc[64+c]
__global__ void k_bnfin(const float* __restrict__ sums,
                        const float* __restrict__ w, const float* __restrict__ b,
                        float invn, float* __restrict__ ksc) {
  int c = threadIdx.x;  // 64
  float m   = sums[c] * invn;
  float var = sums[64 + c] * invn - m * m;
  float sc  = rsqrtf(var + EPS) * w[c];
  ksc[c]      = sc;
  ksc[64 + c] = b[c] - m * sc;
}

__global__ void k_bnapply_prelu(float* __restrict__ x,
                                const float* __restrict__ ksc,
                                const float* __restrict__ a, int n64) {
  int i = blockIdx.x * blockDim.x + threadIdx.x;
  if (i >= n64) return;
  int c = i & 63;
  float y  = fmaf(x[i], ksc[c], ksc[64 + c]);
  float al = a[0];
  x[i] = y >= 0.f ? y : al * y;
}

// ----------------------------------------------- global max pool (per graph)
__device__ __forceinline__ unsigned enc_f32(float f) {
  unsigned u = __float_as_uint(f);
  return (u & 0x80000000u) ? ~u : (u | 0x80000000u);  // order-preserving map
}
__device__ __forceinline__ float dec_f32(unsigned u) {
  return (u & 0x80000000u) ? __uint_as_float(u & 0x7FFFFFFFu)
                           : __uint_as_float(~u);
}

__global__ void k_gmax(const float* __restrict__ h, const int* __restrict__ batch,
                       unsigned* __restrict__ gU, int n64) {
  int i = blockIdx.x * blockDim.x + threadIdx.x;
  if (i >= n64) return;
  int row = i >> 6, c = i & 63;
  atomicMax(&gU[(size_t)batch[row] * 64 + c], enc_f32(h[i]));
}

__global__ void k_gdec(const unsigned* __restrict__ gU, float* __restrict__ g, int n) {
  int i = blockIdx.x * blockDim.x + threadIdx.x;
  if (i < n) g[i] = dec_f32(gU[i]);
}

// -------------------------------------------------------- head: [NG,64]@[64,1]
__global__ void k_dot64(const float* __restrict__ gA, const float* __restrict__ w,
                        const float* __restrict__ b, float* __restrict__ v, int M) {
  int i = blockIdx.x * blockDim.x + threadIdx.x;
  if (i >= M) return;
  float s = b[0];
#pragma unroll 16
  for (int k = 0; k < 64; ++k) s += gA[(size_t)i * 64 + k] * w[k];
  v[i] = s;
}

// single-feature BN over M rows + PReLU, one block of 512 threads
__global__ void k_bn1feat_prelu(const float* __restrict__ v,
                                const float* __restrict__ w, const float* __restrict__ b,
                                const float* __restrict__ a,
                                float* __restrict__ out, int M) {
  __shared__ float ss[512], sq[512];
  int t = threadIdx.x;
  float x = (t < M) ? v[t] : 0.f;
  ss[t] = x; sq[t] = x * x;
  __syncthreads();
  for (int off = 256; off > 0; off >>= 1) {
    if (t < off) { ss[t] += ss[t + off]; sq[t] += sq[t + off]; }
    __syncthreads();
  }
  float m   = ss[0] / (float)M;
  float var = sq[0] / (float)M - m * m;
  float inv = rsqrtf(var + EPS);
  if (t < M) {
    float y = (x - m) * inv * w[0] + b[0];
    out[t] = y >= 0.f ? y : a[0] * y;
  }
}

// ---------------------------------------------------------------------------
extern "C" void kernel_launch(void* const* d_in, const int* in_sizes, int n_in,
                              void* d_out, int out_size, void* d_ws, size_t ws_size,
                              hipStream_t stream) {
  const float* x     = (const float*)d_in[0];
  const int*   eidx  = (const int*)d_in[1];
  const int*   batch = (const int*)d_in[2];
  const float* W1    = (const float*)d_in[3];
  const float* b1    = (const float*)d_in[4];
  const float* W2    = (const float*)d_in[5];
  const float* b2    = (const float*)d_in[6];
  const float* bn1w  = (const float*)d_in[7];
  const float* bn1b  = (const float*)d_in[8];
  const float* bn2w  = (const float*)d_in[9];
  const float* bn2b  = (const float*)d_in[10];
  const float* bn3w  = (const float*)d_in[11];
  const float* bn3b  = (const float*)d_in[12];
  const float* bn4w  = (const float*)d_in[13];
  const float* bn4b  = (const float*)d_in[14];
  const float* lw1   = (const float*)d_in[15];
  const float* lb1   = (const float*)d_in[16];  // constant shift: cancelled by BN3
  const float* lw2   = (const float*)d_in[17];
  const float* lb2   = (const float*)d_in[18];
  const float* a1    = (const float*)d_in[19];
  const float* a2    = (const float*)d_in[20];
  const float* a3    = (const float*)d_in[21];
  const float* a4    = (const float*)d_in[22];
  (void)lb1; (void)n_in; (void)ws_size;

  const int N  = in_sizes[0] / 16;   // 100000 nodes
  const int E  = in_sizes[1] / 2;    // 1250000 edges
  const int NG = out_size;           // 512 graphs
  const int FIN = 16, H = 64;
  const int n64 = N * H;
  const int g64 = NG * H;

  const int* src = eidx;
  const int* dst = eidx + E;

  // ---- workspace carve-up (256B aligned)
  char* ws = (char*)d_ws;
  size_t off = 0;
  auto carve = [&](size_t bytes) {
    size_t o = off;
    off = (off + bytes + 255) & ~(size_t)255;
    return o;
  };
  float*    bufA = (float*)(ws + carve((size_t)n64 * 4));  // agg1 / hA / agg2
  float*    bufB = (float*)(ws + carve((size_t)n64 * 4));  // h1 / h2
  float*    deg  = (float*)(ws + carve((size_t)N * 4));
  float*    dis  = (float*)(ws + carve((size_t)N * 4));
  float*    sums = (float*)(ws + carve(128 * 4));
  float*    ksc  = (float*)(ws + carve(128 * 4));
  unsigned* gU   = (unsigned*)(ws + carve((size_t)g64 * 4));
  float*    g    = (float*)(ws + carve((size_t)g64 * 4));
  float*    t1   = (float*)(ws + carve((size_t)g64 * 4));
  float*    vv   = (float*)(ws + carve((size_t)NG * 4));

  const int TB = 256;
  auto blk = [](long long n, int t) { return (unsigned)((n + t - 1) / t); };

  // ---- degrees (dst-side + self loop), dis = deg^-1/2
  k_fill_f32<<<blk(N, TB), TB, 0, stream>>>(deg, 1.0f, N);
  k_deg<<<blk(E, TB), TB, 0, stream>>>(dst, deg, E);
  k_rsqrt<<<blk(N, TB), TB, 0, stream>>>(deg, dis, N);

  // ================= layer 1 =================
  k_gemm_n64<<<N / 16, 32, 0, stream>>>(x, W1, bufB, FIN);                 // h1 = x@W1
  k_selfloop<<<blk(n64, TB), TB, 0, stream>>>(bufB, dis, b1, bufA, n64);   // agg1 init
  k_edge<<<blk((long long)E * 16, TB), TB, 0, stream>>>(src, dst, dis, bufB, bufA, E);
  k_fill_f32<<<1, 128, 0, stream>>>(sums, 0.f, 128);
  k_bnstats<<<256, 256, 0, stream>>>(bufA, N, sums);
  k_bnfin<<<1, 64, 0, stream>>>(sums, bn1w, bn1b, 1.0f / (float)N, ksc);
  k_bnapply_prelu<<<blk(n64, TB), TB, 0, stream>>>(bufA, ksc, a1, n64);    // hA

  // ================= layer 2 =================
  k_gemm_n64<<<N / 16, 32, 0, stream>>>(bufA, W2, bufB, H);                // h2 = hA@W2
  k_selfloop<<<blk(n64, TB), TB, 0, stream>>>(bufB, dis, b2, bufA, n64);   // agg2 init
  k_edge<<<blk((long long)E * 16, TB), TB, 0, stream>>>(src, dst, dis, bufB, bufA, E);
  k_fill_f32<<<1, 128, 0, stream>>>(sums, 0.f, 128);
  k_bnstats<<<256, 256, 0, stream>>>(bufA, N, sums);
  k_bnfin<<<1, 64, 0, stream>>>(sums, bn2w, bn2b, 1.0f / (float)N, ksc);
  k_bnapply_prelu<<<blk(n64, TB), TB, 0, stream>>>(bufA, ksc, a2, n64);    // hB

  // ================= global max pool =================
  k_fill_u32<<<blk(g64, TB), TB, 0, stream>>>(gU, 0x007FFFFFu /* enc(-inf) */, g64);
  k_gmax<<<blk(n64, TB), TB, 0, stream>>>(bufA, batch, gU, n64);
  k_gdec<<<blk(g64, TB), TB, 0, stream>>>(gU, g, g64);

  // ================= MLP head =================
  k_gemm_n64<<<NG / 16, 32, 0, stream>>>(g, lw1, t1, H);                   // t1 = g@lw1
  k_fill_f32<<<1, 128, 0, stream>>>(sums, 0.f, 128);
  k_bnstats<<<16, 256, 0, stream>>>(t1, NG, sums);
  k_bnfin<<<1, 64, 0, stream>>>(sums, bn3w, bn3b, 1.0f / (float)NG, ksc);
  k_bnapply_prelu<<<blk(g64, TB), TB, 0, stream>>>(t1, ksc, a3, g64);      // gA

  k_dot64<<<blk(NG, TB), TB, 0, stream>>>(t1, lw2, lb2, vv, NG);
  k_bn1feat_prelu<<<1, 512, 0, stream>>>(vv, bn4w, bn4b, a4, (float*)d_out, NG);
}